// DataParallel_wrapper_90280212562058
// MI455X (gfx1250) — compile-verified
//
#include <hip/hip_runtime.h>

typedef __attribute__((ext_vector_type(2))) float v2f;
typedef __attribute__((ext_vector_type(8))) float v8f;

static constexpr int kB = 8;
static constexpr int kC = 256;
static constexpr int kN = 2048;
static constexpr int kR = 100000;

// ---------------------------------------------------------------------------
// Phase 1: proj[b][n][j] = sum_c feats[b,c,n] * Wcat[c,j], j=0..7
//   Wcat[:,0:2] = W_row[0:256,:]   Wcat[:,2:4] = W_row[256:512,:]
//   Wcat[:,4:6] = W_col[0:256,:]   Wcat[:,6:8] = W_col[256:512,:]
// One wave per 32-wide n-chunk (two 16-wide M tiles sharing each B fragment).
// K=256 via chained V_WMMA_F32_16X16X4_F32.
// Weights live in LDS transposed+padded: wldsT[col][k], 16 cols (8 real + 8
// zero), stride 260 floats -> unconditional conflict-free ds_load_b64.
// ---------------------------------------------------------------------------
static constexpr int kWS = kC + 4;  // padded K-stride (260) for bank spread

__global__ __launch_bounds__(256) void proj_gemm_kernel(
    const float* __restrict__ feats, const float* __restrict__ W_row,
    const float* __restrict__ W_col, float* __restrict__ proj) {
  __shared__ float wldsT[16 * kWS];
  for (int t = threadIdx.x; t < 16 * kWS; t += 256) {
    int col = t / kWS, k = t % kWS;
    float v = 0.0f;
    if (col < 8 && k < kC) {
      if (col < 2)      v = W_row[k * 2 + col];
      else if (col < 4) v = W_row[(kC + k) * 2 + (col - 2)];
      else if (col < 6) v = W_col[k * 2 + (col - 4)];
      else              v = W_col[(kC + k) * 2 + (col - 6)];
    }
    wldsT[t] = v;
  }
  __syncthreads();

  const int lane = threadIdx.x & 31;
  const int wave = threadIdx.x >> 5;
  const int id   = blockIdx.x * 8 + wave;  // 0..511: 32-wide n-chunk
  const int gb   = id >> 6;                // batch index 0..7
  const int n0   = (id & 63) << 5;         // chunk start in n
  const int m    = lane & 15;              // A: row M / B+D: column
  const int koff = (lane >> 4) << 1;       // K sub-offset: 0 | 2

  const float* fb = feats + (size_t)gb * kC * kN + n0 + m;
  const float* wp = &wldsT[m * kWS + koff];

  v8f acc0 = {}, acc1 = {};
  for (int kk = 0; kk < kC; kk += 4) {
    const float* f0 = fb + (size_t)(kk + koff) * kN;
    v2f a0, a1, w;
    a0.x = f0[0];        // A tile0: feats[gb][kk+koff][n0+m]
    a0.y = f0[kN];       //          feats[gb][kk+koff+1][n0+m]
    a1.x = f0[16];       // A tile1: n0+16+m
    a1.y = f0[kN + 16];
    w = *(const v2f*)(wp + kk);  // B[K=kk+koff..+1][col=m], single b64
    acc0 = __builtin_amdgcn_wmma_f32_16x16x4_f32(false, a0, false, w,
                                                 (short)0, acc0, false, false);
    acc1 = __builtin_amdgcn_wmma_f32_16x16x4_f32(false, a1, false, w,
                                                 (short)0, acc1, false, false);
  }

  // D layout: VGPR j -> M = j + (lane>=16 ? 8 : 0), N = lane&15
  if (m < 8) {
    const int mhi = (lane >> 4) << 3;
    const size_t rowBase = (size_t)gb * kN + n0 + mhi;
#pragma unroll
    for (int j = 0; j < 8; ++j) {
      proj[(rowBase + j) * 8 + m]      = acc0[j];
      proj[(rowBase + 16 + j) * 8 + m] = acc1[j];
    }
  }
}

// ---------------------------------------------------------------------------
// Phase 2: per-sample loss.  logits -> softmax -> log_softmax(probs) -> NLL.
// ---------------------------------------------------------------------------
__device__ __forceinline__ float pair_loss(float l0, float l1, bool gt1) {
  float mx  = fmaxf(l0, l1);
  float e0  = expf(l0 - mx), e1 = expf(l1 - mx);
  float inv = 1.0f / (e0 + e1);
  float p0  = e0 * inv, p1 = e1 * inv;
  // log_softmax applied to the *probs* (as in the reference)
  float m2  = fmaxf(p0, p1);
  float lse = m2 + logf(expf(p0 - m2) + expf(p1 - m2));
  return lse - (gt1 ? p1 : p0);  // = -(p[gt] - lse)
}

__global__ __launch_bounds__(256) void loss_kernel(
    const float* __restrict__ proj, const int* __restrict__ pb,
    const int* __restrict__ pi, const int* __restrict__ pj,
    const int* __restrict__ nb, const int* __restrict__ ni,
    const float* __restrict__ bias, int projOff, float* __restrict__ accum) {
  int r = blockIdx.x * 256 + threadIdx.x;
  float s = 0.0f;
  if (r < kR) {
    size_t basePI = ((size_t)pb[r] * kN + pi[r]) * 8 + projOff;
    size_t basePJ = ((size_t)pb[r] * kN + pj[r]) * 8 + projOff;
    size_t baseNI = ((size_t)nb[r] * kN + ni[r]) * 8 + projOff;
    float4 q  = *(const float4*)(proj + basePI);      // A(pb,pi), B(pb,pi)
    float2 bj = *(const float2*)(proj + basePJ + 2);  // B(pb,pj)
    float2 na = *(const float2*)(proj + baseNI);      // A(nb,ni)
    float b0 = bias[0], b1 = bias[1];
    s += pair_loss(q.x  + bj.x + b0, q.y  + bj.y + b1, true);   // pos
    s += pair_loss(na.x + q.z  + b0, na.y + q.w  + b1, false);  // neg w/ p1
    s += pair_loss(na.x + bj.x + b0, na.y + bj.y + b1, false);  // neg w/ p2
  }
  // wave32 butterfly reduction
#pragma unroll
  for (int off = 16; off > 0; off >>= 1) s += __shfl_xor(s, off, 32);
  __shared__ float wsum[8];
  if ((threadIdx.x & 31) == 0) wsum[threadIdx.x >> 5] = s;
  __syncthreads();
  if (threadIdx.x == 0) {
    float t = 0.0f;
#pragma unroll
    for (int w = 0; w < 8; ++w) t += wsum[w];
    atomicAdd(accum, t);
  }
}

__global__ void init_kernel(float* accum) { *accum = 0.0f; }

__global__ void final_kernel(const float* __restrict__ accum,
                             float* __restrict__ out) {
  out[0] = accum[0] / (3.0f * (float)kR);
}

// ---------------------------------------------------------------------------
extern "C" void kernel_launch(void* const* d_in, const int* in_sizes, int n_in,
                              void* d_out, int out_size, void* d_ws,
                              size_t ws_size, hipStream_t stream) {
  const float* feats = (const float*)d_in[0];
  const float* W_row = (const float*)d_in[1];
  const float* b_row = (const float*)d_in[2];
  const float* W_col = (const float*)d_in[3];
  const float* b_col = (const float*)d_in[4];
  const int* row_pos_b = (const int*)d_in[5];
  const int* row_pos_i = (const int*)d_in[6];
  const int* row_pos_j = (const int*)d_in[7];
  const int* row_neg_b = (const int*)d_in[8];
  const int* row_neg_i = (const int*)d_in[9];
  const int* col_pos_b = (const int*)d_in[10];
  const int* col_pos_i = (const int*)d_in[11];
  const int* col_pos_j = (const int*)d_in[12];
  const int* col_neg_b = (const int*)d_in[13];
  const int* col_neg_i = (const int*)d_in[14];
  // d_in[15]/d_in[16] (row_perm/col_perm): permutation of a mean -> no-op.

  float* accum = (float*)d_ws;                  // 1 float
  float* proj  = (float*)((char*)d_ws + 256);   // 16384*8 floats = 512 KB

  init_kernel<<<1, 1, 0, stream>>>(accum);
  proj_gemm_kernel<<<64, 256, 0, stream>>>(feats, W_row, W_col, proj);

  const int lb = (kR + 255) / 256;
  loss_kernel<<<lb, 256, 0, stream>>>(proj, row_pos_b, row_pos_i, row_pos_j,
                                      row_neg_b, row_neg_i, b_row, 0, accum);
  loss_kernel<<<lb, 256, 0, stream>>>(proj, col_pos_b, col_pos_i, col_pos_j,
                                      col_neg_b, col_neg_i, b_col, 4, accum);
  final_kernel<<<1, 1, 0, stream>>>(accum, (float*)d_out);
}